// Generator_38903813767219
// MI455X (gfx1250) — compile-verified
//
#include <hip/hip_runtime.h>

typedef __attribute__((ext_vector_type(2))) float v2f;
typedef __attribute__((ext_vector_type(8))) float v8f;

#define NB      2048
#define FC_K    100
#define FC_N    17640          // 5*98*36
#define NTILES  1103           // ceil(17640/16), last tile 8 valid cols
#define NGRPS   138            // ceil(1103/8)
#define H1_ELEMS (NB * FC_N)                  // 36,126,720
#define H2_ELEMS (NB * 5 * 98 * 16)           // 16,056,320
#define CONV_BLOCKS 7840                      // 62720 waves / 8
#define PART_ELEMS (CONV_BLOCKS * 32)         // 250,880
#define BN_COUNT 1003520.0f                   // 2048*5*98

// ---------------------------------------------------------------------------
// Kernel 1: FC GEMM  h1[b, (i*98+j)*36+c] = x @ fc_w^T + fc_b
// One wave = one 16-row M tile x 8 consecutive 16-col N tiles, K=100 via
// 25 x v_wmma_f32_16x16x4_f32. A fragments loaded once, reused per N tile.
// Fast path (no guards) for the 1102 full tiles; guarded path only for the
// final partial tile -> no cndmask in the hot load/WMMA chain.
// ---------------------------------------------------------------------------
__global__ void __launch_bounds__(256) fc_wmma(const float* __restrict__ x,
                                               const float* __restrict__ fw,
                                               const float* __restrict__ fb,
                                               float* __restrict__ h1) {
  const int lane  = threadIdx.x & 31;
  const int wave  = threadIdx.x >> 5;
  const int w     = blockIdx.x * 8 + wave;
  const int mtile = w / NGRPS;
  const int ngrp  = w - mtile * NGRPS;
  const int lrow  = lane & 15;
  const int lhalf = lane >> 4;
  const int row   = mtile * 16 + lrow;
  const int kb0   = 2 * lhalf;

  v2f a[25];
#pragma unroll
  for (int s = 0; s < 25; ++s)
    a[s] = *(const v2f*)(x + row * FC_K + 4 * s + kb0);

  for (int t = 0; t < 8; ++t) {
    int ntile = ngrp * 8 + t;
    if (ntile >= NTILES) break;
    int n = ntile * 16 + lrow;
    v8f acc = {};
    if (ntile < NTILES - 1) {
      // ---- fast path: all 16 columns valid ----
      const float* bp = fw + n * FC_K + kb0;
#pragma unroll
      for (int s = 0; s < 25; ++s) {
        v2f bf = *(const v2f*)(bp + 4 * s);
        acc = __builtin_amdgcn_wmma_f32_16x16x4_f32(false, a[s], false, bf,
                                                    (short)0, acc, false, false);
      }
      float bias = fb[n];
#pragma unroll
      for (int v = 0; v < 8; ++v) {
        int m = mtile * 16 + lhalf * 8 + v;
        h1[m * FC_N + n] = acc[v] + bias;
      }
    } else {
      // ---- tail path: last tile, only 8 of 16 columns valid ----
      bool nvalid = (n < FC_N);
      const float* bp = fw + (nvalid ? n : 0) * FC_K + kb0;
#pragma unroll
      for (int s = 0; s < 25; ++s) {
        v2f bf = *(const v2f*)(bp + 4 * s);
        if (!nvalid) { bf[0] = 0.f; bf[1] = 0.f; }
        acc = __builtin_amdgcn_wmma_f32_16x16x4_f32(false, a[s], false, bf,
                                                    (short)0, acc, false, false);
      }
      float bias = nvalid ? fb[n] : 0.f;
#pragma unroll
      for (int v = 0; v < 8; ++v) {
        int m = mtile * 16 + lhalf * 8 + v;
        if (nvalid) h1[m * FC_N + n] = acc[v] + bias;
      }
    }
  }
}

// ---------------------------------------------------------------------------
// Kernel 1b: one-shot repack of conv1 weights [16][36][2][2] into
// pw[n][k] with k = p*36 + c (p = kh*2+kw), so conv B-fragments are a single
// contiguous 64-bit load per k-step.
// ---------------------------------------------------------------------------
__global__ void pack_conv_w(const float* __restrict__ cw, float* __restrict__ pw) {
  int idx = blockIdx.x * 256 + threadIdx.x;   // 2304 = 16*144 total
  if (idx >= 2304) return;
  int n = idx / 144;
  int k = idx - n * 144;
  int p = k / 36;
  int c = k - p * 36;
  pw[idx] = cw[n * 144 + c * 4 + p];
}

// ---------------------------------------------------------------------------
// Kernel 2: conv1(2x2, pad bottom/right) + bias + leaky, as per-location GEMM
// [16 batch x 144] @ [144 x 16ch]. Patch index p is wave-uniform per k-step
// (boundaries 36/72/108 are 4-aligned) and compile-time constant under full
// unroll -> padding check is a uniform scalar branch, EXEC stays all-1s for
// every WMMA. Emits deterministic per-block BN partials (sum, sumsq).
// h2 layout: [b][i][j][c16].
// ---------------------------------------------------------------------------
__global__ void __launch_bounds__(256) conv_wmma(const float* __restrict__ h1,
                                                 const float* __restrict__ pw,
                                                 const float* __restrict__ cb,
                                                 float* __restrict__ h2,
                                                 float* __restrict__ part) {
  __shared__ float wsum[8][16];
  __shared__ float wsq[8][16];
  const int lane  = threadIdx.x & 31;
  const int wave  = threadIdx.x >> 5;
  const int w     = blockIdx.x * 8 + wave;
  const int mtile = w / 490;
  const int sp    = w - mtile * 490;
  const int i     = sp / 98;
  const int j     = sp - i * 98;
  const int lrow  = lane & 15;
  const int lhalf = lane >> 4;
  const int bb    = mtile * 16 + lrow;
  const float* bwp = pw + lrow * 144 + 2 * lhalf;

  v8f acc = {};
#pragma unroll
  for (int s = 0; s < 36; ++s) {
    const int p  = (4 * s) / 36;           // compile-time: 0,0,..,1,..,2,..,3
    const int di = p >> 1, dj = p & 1;
    const int ii = i + di, jj = j + dj;    // wave-uniform
    int c0 = 4 * s - p * 36 + 2 * lhalf;
    v2f af;
    if (ii < 5 && jj < 98) {               // uniform scalar branch
      af = *(const v2f*)(h1 + (((bb * 5 + ii) * 98 + jj) * 36 + c0));
    } else {
      af[0] = 0.f; af[1] = 0.f;            // zero padding (bottom/right)
    }
    v2f bf = *(const v2f*)(bwp + 4 * s);
    acc = __builtin_amdgcn_wmma_f32_16x16x4_f32(false, af, false, bf,
                                                (short)0, acc, false, false);
  }

  float bias = cb[lrow];
  float s0 = 0.f, q0 = 0.f;
#pragma unroll
  for (int v = 0; v < 8; ++v) {
    int m = mtile * 16 + lhalf * 8 + v;
    float val = acc[v] + bias;
    val = val >= 0.f ? val : 0.01f * val;
    h2[((m * 5 + i) * 98 + j) * 16 + lrow] = val;
    s0 += val;
    q0 += val * val;
  }
  // combine the two lane-halves (same channel, other 8 batch rows)
  s0 += __shfl_xor(s0, 16, 32);
  q0 += __shfl_xor(q0, 16, 32);
  if (lane < 16) { wsum[wave][lane] = s0; wsq[wave][lane] = q0; }
  __syncthreads();
  if (threadIdx.x < 32) {
    int c = threadIdx.x & 15;
    int st = threadIdx.x >> 4;
    float tot = 0.f;
    for (int wv = 0; wv < 8; ++wv) tot += st ? wsq[wv][c] : wsum[wv][c];
    part[blockIdx.x * 32 + threadIdx.x] = tot;  // [block][stat*16+c]
  }
}

// ---------------------------------------------------------------------------
// Kernel 3: deterministic BN finalize -> per-channel scale/shift
// ---------------------------------------------------------------------------
__global__ void __launch_bounds__(256) bn_finalize(const float* __restrict__ part,
                                                   const float* __restrict__ gamma,
                                                   const float* __restrict__ beta,
                                                   float* __restrict__ scale,
                                                   float* __restrict__ shift) {
  __shared__ float acc[256];
  __shared__ float tot[32];
  int t = threadIdx.x;
  int cs = t & 31, r = t >> 5;
  float s = 0.f;
  for (int blk = r; blk < CONV_BLOCKS; blk += 8) s += part[blk * 32 + cs];
  acc[t] = s;
  __syncthreads();
  if (t < 32) {
    float sum = 0.f;
    for (int rr = 0; rr < 8; ++rr) sum += acc[rr * 32 + t];
    tot[t] = sum;
  }
  __syncthreads();
  if (t < 16) {
    const float invN = 1.0f / BN_COUNT;
    float mean = tot[t] * invN;
    float var  = tot[16 + t] * invN - mean * mean;
    float inv  = rsqrtf(var + 1e-5f);
    float scl  = gamma[t] * inv;
    scale[t] = scl;
    shift[t] = beta[t] - mean * scl;
  }
}

// ---------------------------------------------------------------------------
// Kernel 4: BN apply + lc1 (+leaky) + lc2, one batch element per block.
// h2[b] (5x98x16 = 31KB) staged in LDS via b128 loads with BN applied
// (channel group per thread is fixed since 128 % 16 == 0); h3 (6x4x97) in LDS.
// ---------------------------------------------------------------------------
#define H2PB 7840
#define H3SZ 2328
__global__ void __launch_bounds__(128) lc_kernel(const float* __restrict__ h2,
                                                 const float* __restrict__ scale,
                                                 const float* __restrict__ shift,
                                                 const float* __restrict__ w1,
                                                 const float* __restrict__ b1,
                                                 const float* __restrict__ w2,
                                                 const float* __restrict__ b2,
                                                 float* __restrict__ out) {
  __shared__ float sh2[H2PB];
  __shared__ float sh3[H3SZ];
  const int b = blockIdx.x;
  const int tid = threadIdx.x;

  // BN-apply while staging h2[b] into LDS, 128-bit at a time.
  const float4 sc4 = ((const float4*)scale)[tid & 3];  // channels 4q..4q+3
  const float4 sf4 = ((const float4*)shift)[tid & 3];
  const float4* src4 = (const float4*)(h2 + b * H2PB);
  for (int idx = tid; idx < H2PB / 4; idx += 128) {
    float4 v = src4[idx];
    v.x = v.x * sc4.x + sf4.x;
    v.y = v.y * sc4.y + sf4.y;
    v.z = v.z * sc4.z + sf4.z;
    v.w = v.w * sc4.w + sf4.w;
    ((float4*)sh2)[idx] = v;
  }
  __syncthreads();

  for (int e = tid; e < H3SZ; e += 128) {
    int o1 = e / 388;
    int r  = e - o1 * 388;
    int oi = r / 97;
    int oj = r - oi * 97;
    float acc = b1[e];  // b1 flat == o1*388 + oi*97 + oj
    const float* wp = w1 + o1 * 24832 + oi * 388 + oj * 4;  // c stride 1552
    const float* xb = sh2 + (oi * 98 + oj) * 16;
#pragma unroll
    for (int c = 0; c < 16; ++c) {
      float4 wv = *(const float4*)(wp + c * 1552);
      acc += wv.x * xb[c]            // (oi,   oj)
           + wv.y * xb[16 + c]       // (oi,   oj+1)
           + wv.z * xb[98 * 16 + c]  // (oi+1, oj)
           + wv.w * xb[99 * 16 + c]; // (oi+1, oj+1)
    }
    acc = acc >= 0.f ? acc : 0.01f * acc;
    sh3[e] = acc;
  }
  __syncthreads();

  for (int e = tid; e < 288; e += 128) {
    int ri = e / 96;
    int rj = e - ri * 96;
    float acc = b2[e];
#pragma unroll
    for (int o1 = 0; o1 < 6; ++o1) {
      float4 wv = *(const float4*)(w2 + (((o1 * 3 + ri) * 96 + rj) << 2));
      const float* xb = sh3 + o1 * 388 + ri * 97 + rj;
      acc += wv.x * xb[0] + wv.y * xb[1] + wv.z * xb[97] + wv.w * xb[98];
    }
    out[b * 288 + e] = acc;
  }
}

// ---------------------------------------------------------------------------
extern "C" void kernel_launch(void* const* d_in, const int* in_sizes, int n_in,
                              void* d_out, int out_size, void* d_ws, size_t ws_size,
                              hipStream_t stream) {
  const float* x     = (const float*)d_in[0];
  const float* fw    = (const float*)d_in[1];
  const float* fb    = (const float*)d_in[2];
  const float* cw    = (const float*)d_in[3];
  const float* cb    = (const float*)d_in[4];
  const float* gamma = (const float*)d_in[5];
  const float* beta  = (const float*)d_in[6];
  const float* w1    = (const float*)d_in[7];
  const float* b1    = (const float*)d_in[8];
  const float* w2    = (const float*)d_in[9];
  const float* b2    = (const float*)d_in[10];

  float* ws    = (float*)d_ws;
  float* h1    = ws;                               // 36,126,720 f
  float* h2    = ws + H1_ELEMS;                    // 16,056,320 f
  float* part  = ws + H1_ELEMS + H2_ELEMS;         // 250,880 f
  float* scale = part + PART_ELEMS;                // 16 f (16B aligned)
  float* shift = scale + 16;                       // 16 f
  float* pw    = shift + 16;                       // 2,304 f packed conv1 w

  pack_conv_w<<<9, 256, 0, stream>>>(cw, pw);
  fc_wmma    <<<2208, 256, 0, stream>>>(x, fw, fb, h1);
  conv_wmma  <<<CONV_BLOCKS, 256, 0, stream>>>(h1, pw, cb, h2, part);
  bn_finalize<<<1, 256, 0, stream>>>(part, gamma, beta, scale, shift);
  lc_kernel  <<<NB, 128, 0, stream>>>(h2, scale, shift, w1, b1, w2, b2,
                                      (float*)d_out);
}